// WTConv2d_Fused_154618822866
// MI455X (gfx1250) — compile-verified
//
#include <hip/hip_runtime.h>
#include <hip/hip_bf16.h>

typedef __attribute__((ext_vector_type(2))) float v2f;
typedef __attribute__((ext_vector_type(8))) float v8f;

// ---------------------------------------------------------------------------
// Problem constants (from reference setup_inputs): B=8, C=32, H=W=224,
// weight (128,128,3,3), 3 wavelet levels. 224 % 8 == 0 -> no reflect pad.
// ---------------------------------------------------------------------------
#define BATCH 8
#define CBASE 32      // base channels
#define CFULL 128     // CBASE*4 (subband-expanded channels)
#define KTOT  1152    // CFULL * 3 * 3
#define KC    36      // K tile (4 cin * 9 taps); 1152 % 36 == 0
#define WPAD  44      // LDS row stride (dwords) for weight tile:
                      //   44*l mod 64 distinct for l=0..15 (conflict-free),
                      //   44 % 4 == 0 (keeps b128 LDS stores aligned)
#define BPAD  34      // LDS row stride (dwords) for im2col tile

// ---------------------------------------------------------------------------
// Haar DWT: read (2h x 2w) plane, write 4 subbands into (B,128,h,w) tensor
// with channel layout c*4 + band (matches sub.reshape(B, C*4, h, w)).
// Input may be a strided view (band-0 channels of the previous level's sub).
// Templated on output dims so index math uses const-division (no v_rcp).
// ---------------------------------------------------------------------------
template <int Hh, int Ww>
__global__ __launch_bounds__(256) void wt_dwt_kernel(
    const float* __restrict__ in, float* __restrict__ out,
    int inBS, int inCS, int total)
{
  int i = blockIdx.x * 256 + threadIdx.x;
  if (i >= total) return;
  int x = i % Ww; int t = i / Ww;
  int y = t % Hh; t /= Hh;
  int c = t % CBASE; int b = t / CBASE;

  const float* p = in + (size_t)b * inBS + (size_t)c * inCS
                      + (size_t)(2 * y) * (2 * Ww) + 2 * x;
  float a = p[0], bb = p[1], cc = p[2 * Ww], dd = p[2 * Ww + 1];

  float ll = (a + bb + cc + dd) * 0.5f;
  float lh = (a - bb + cc - dd) * 0.5f;
  float hl = (a + bb - cc - dd) * 0.5f;
  float hh = (a - bb - cc + dd) * 0.5f;

  const int hw = Hh * Ww;
  float* q = out + (size_t)(b * CFULL + c * 4) * hw + y * Ww + x;
  q[0]      = ll;
  q[hw]     = lh;
  q[2 * hw] = hl;
  q[3 * hw] = hh;
}

// ---------------------------------------------------------------------------
// Implicit-GEMM 3x3 conv (pad=1, stride=1) using V_WMMA_F32_16X16X4_F32.
//   D[cout, n] = sum_k  W[cout, k] * im2col[k, n],  k = cin*9 + ky*3 + kx
// Block: 256 threads = 8 waves. wave = (wc in 0..3 | wsp in 0..1):
//   wave computes couts [wc*32, wc*32+32) x spatial cols [wsp*16, wsp*16+16)
// Block tile: 128 couts x 32 spatial columns. Grid.x = (B*h*w)/32 (exact).
// ---------------------------------------------------------------------------
template <int Hh, int Ww>
__global__ __launch_bounds__(256) void wt_conv_wmma_kernel(
    const float* __restrict__ A,   // subbands (B,128,h,w)
    const float* __restrict__ Wt,  // weight  (128,128,3,3) = [cout][k]
    float* __restrict__ Y)         // output  (B,128,h,w)
{
  __shared__ float sW[CFULL * WPAD]; // weight tile  [128][KC]
  __shared__ float sB[KC * BPAD];    // im2col tile  [KC][32]

  constexpr int hw = Hh * Ww;
  const int tid  = threadIdx.x;
  const int wave = tid >> 5;
  const int lane = tid & 31;
  const int half = lane >> 4;
  const int lr   = lane & 15;
  const int wc   = wave & 3;   // cout group
  const int wsp  = wave >> 2;  // spatial half
  const int nbase = blockIdx.x * 32;

  // Per-thread fixed im2col column (256 % 32 == 0 -> nl invariant): decode once.
  const int nlfix = tid & 31;
  const int kwave = tid >> 5;                  // staging row start (0..7)
  {
  }
  const int ngf = nbase + nlfix;
  const int bF  = ngf / hw;  const int sF  = ngf - bF * hw;
  const int oyF = sF / Ww;   const int oxF = sF - oyF * Ww;
  const float* Abase = A + (size_t)bF * CFULL * hw;

  v8f acc0 = {};
  v8f acc1 = {};

  for (int kt = 0; kt < KTOT; kt += KC) {
    __syncthreads();
    // Stage weight tile: 128 rows x 9 float4 (b128 global + b128 LDS stores).
    for (int e = tid; e < CFULL * (KC / 4); e += 256) {
      int co = e / 9; int j4 = e - co * 9;
      float4 vv = *(const float4*)(Wt + (size_t)co * KTOT + kt + j4 * 4);
      *(float4*)(&sW[co * WPAD + j4 * 4]) = vv;
    }
    // Stage im2col tile: each thread owns column nlfix, rows kwave + 8*j.
    for (int kk = kwave; kk < KC; kk += 8) {
      int k  = kt + kk;
      int cin = k / 9; int r = k - cin * 9;
      int ky = r / 3;  int kx = r - ky * 3;
      int iy = oyF + ky - 1;
      int ix = oxF + kx - 1;
      float v = 0.0f;
      if ((unsigned)iy < (unsigned)Hh && (unsigned)ix < (unsigned)Ww)
        v = Abase[(size_t)cin * hw + iy * Ww + ix];
      sB[kk * BPAD + nlfix] = v;
    }
    // Prefetch next weight K-tile into cache while we compute this one
    // (lowers to global_prefetch_b8; L2-resident weight, cheap and safe).
    if (kt + KC < KTOT) {
      int row = tid >> 1, part = tid & 1;
      __builtin_prefetch(Wt + (size_t)row * KTOT + (kt + KC) + part * 18, 0, 1);
    }
    __syncthreads();

    const int ct0 = wc * 32;
    const int nl  = wsp * 16 + lr;
#pragma unroll
    for (int kk = 0; kk < KC; kk += 4) {
      const int ko = kk + 2 * half;  // this lane-half holds K = ko, ko+1
      v2f bf, a0, a1;
      bf.x = sB[ko * BPAD + nl];
      bf.y = sB[(ko + 1) * BPAD + nl];
      a0.x = sW[(ct0 + lr) * WPAD + ko];
      a0.y = sW[(ct0 + lr) * WPAD + ko + 1];
      a1.x = sW[(ct0 + 16 + lr) * WPAD + ko];
      a1.y = sW[(ct0 + 16 + lr) * WPAD + ko + 1];
      acc0 = __builtin_amdgcn_wmma_f32_16x16x4_f32(
          false, a0, false, bf, (short)0, acc0, false, false);
      acc1 = __builtin_amdgcn_wmma_f32_16x16x4_f32(
          false, a1, false, bf, (short)0, acc1, false, false);
    }
  }

  // Store: lane's column ng -> consecutive spatial addresses across lanes.
  // D layout: VGPR r -> M = r (lanes 0-15) / r+8 (lanes 16-31).
  int ng = nbase + wsp * 16 + lr;
  int b  = ng / hw; int s = ng - b * hw;
  float* yb = Y + (size_t)b * CFULL * hw + s;
  int m0 = wc * 32 + half * 8;
#pragma unroll
  for (int r = 0; r < 8; ++r) {
    yb[(size_t)(m0 + r) * hw]      = acc0[r];
    yb[(size_t)(m0 + 16 + r) * hw] = acc1[r];
  }
}

// ---------------------------------------------------------------------------
// Inverse Haar: ll = Y[band0] (+ carried next_ll), reconstruct (2h x 2w).
// Final level adds bias and writes to d_out.
// ---------------------------------------------------------------------------
template <int Hh, int Ww>
__global__ __launch_bounds__(256) void wt_idwt_kernel(
    const float* __restrict__ Yl, const float* __restrict__ nll,
    const float* __restrict__ bias, float* __restrict__ out, int total)
{
  int i = blockIdx.x * 256 + threadIdx.x;
  if (i >= total) return;
  int x = i % Ww; int t = i / Ww;
  int y = t % Hh; t /= Hh;
  int c = t % CBASE; int b = t / CBASE;

  constexpr int hw = Hh * Ww;
  const float* p = Yl + (size_t)(b * CFULL + c * 4) * hw + y * Ww + x;
  float ll = p[0];
  if (nll) ll += nll[(size_t)(b * CBASE + c) * hw + y * Ww + x];
  float lh = p[hw], hl = p[2 * hw], hh = p[3 * hw];

  float av = (ll + lh + hl + hh) * 0.5f;
  float bv = (ll - lh + hl - hh) * 0.5f;
  float cv = (ll + lh - hl - hh) * 0.5f;
  float dv = (ll - lh - hl + hh) * 0.5f;

  float bi = bias ? bias[c] : 0.0f;
  float* q = out + (size_t)(b * CBASE + c) * (4 * hw)
                 + (size_t)(2 * y) * (2 * Ww) + 2 * x;
  q[0]          = av + bi;
  q[1]          = bv + bi;
  q[2 * Ww]     = cv + bi;
  q[2 * Ww + 1] = dv + bi;
}

// ---------------------------------------------------------------------------
// Host-side orchestration.
// ---------------------------------------------------------------------------
extern "C" void kernel_launch(void* const* d_in, const int* in_sizes, int n_in,
                              void* d_out, int out_size, void* d_ws, size_t ws_size,
                              hipStream_t stream) {
  const float* x      = (const float*)d_in[0];
  const float* weight = (const float*)d_in[1];
  const float* bias   = (const float*)d_in[2];
  float*       out    = (float*)d_out;
  float*       ws     = (float*)d_ws;

  const int H = 224, W = 224;
  const int h0 = 112, w0 = 112, h1 = 56, w1 = 56, h2 = 28, w2 = 28;

  const size_t SUB0 = (size_t)BATCH * CFULL * h0 * w0; // 12,845,056
  const size_t SUB1 = (size_t)BATCH * CFULL * h1 * w1; //  3,211,264
  const size_t SUB2 = (size_t)BATCH * CFULL * h2 * w2; //    802,816
  const size_t LL1  = (size_t)BATCH * CBASE * h1 * w1;

  float* sub0 = ws;
  float* sub1 = sub0 + SUB0;
  float* sub2 = sub1 + SUB1;
  float* y0   = sub2 + SUB2;
  float* y1   = y0 + SUB0;
  float* y2   = y1 + SUB1;
  float* t2   = y2 + SUB2;        // next_ll after level-2 idwt (B,32,56,56)
  float* t1   = t2 + LL1;         // next_ll after level-1 idwt (B,32,112,112)
  (void)ws_size; (void)in_sizes; (void)n_in; (void)out_size;

  // ---- Forward DWT cascade -------------------------------------------------
  {
    int total = BATCH * CBASE * h0 * w0;
    wt_dwt_kernel<112, 112><<<(total + 255) / 256, 256, 0, stream>>>(
        x, sub0, CBASE * H * W, H * W, total);
  }
  {
    int total = BATCH * CBASE * h1 * w1;
    wt_dwt_kernel<56, 56><<<(total + 255) / 256, 256, 0, stream>>>(
        sub0, sub1, CFULL * h0 * w0, 4 * h0 * w0, total);
  }
  {
    int total = BATCH * CBASE * h2 * w2;
    wt_dwt_kernel<28, 28><<<(total + 255) / 256, 256, 0, stream>>>(
        sub1, sub2, CFULL * h1 * w1, 4 * h1 * w1, total);
  }

  // ---- WMMA convolutions ---------------------------------------------------
  wt_conv_wmma_kernel<112, 112><<<(BATCH * h0 * w0) / 32, 256, 0, stream>>>(sub0, weight, y0);
  wt_conv_wmma_kernel<56, 56><<<(BATCH * h1 * w1) / 32, 256, 0, stream>>>(sub1, weight, y1);
  wt_conv_wmma_kernel<28, 28><<<(BATCH * h2 * w2) / 32, 256, 0, stream>>>(sub2, weight, y2);

  // ---- Inverse DWT cascade with LL accumulation ----------------------------
  {
    int total = BATCH * CBASE * h2 * w2;
    wt_idwt_kernel<28, 28><<<(total + 255) / 256, 256, 0, stream>>>(
        y2, nullptr, nullptr, t2, total);
  }
  {
    int total = BATCH * CBASE * h1 * w1;
    wt_idwt_kernel<56, 56><<<(total + 255) / 256, 256, 0, stream>>>(
        y1, t2, nullptr, t1, total);
  }
  {
    int total = BATCH * CBASE * h0 * w0;
    wt_idwt_kernel<112, 112><<<(total + 255) / 256, 256, 0, stream>>>(
        y0, t1, bias, out, total);
  }
}